// GATLayer_adde_60971355734168
// MI455X (gfx1250) — compile-verified
//
#include <hip/hip_runtime.h>
#include <math.h>

// ---------------------------------------------------------------------------
// GATLayer fused pipeline for MI455X (gfx1250, wave32).
//
// Restructuring vs reference: QKV projections of the two node tokens are
// hoisted per-node (zq/zk/zv = h @ (W* @ W_lin)^T via v_wmma_f32_16x16x4_f32,
// all three matrices fused in one kernel sharing A fragments), and the
// edge-token projection uses folded [64,11] weights computed once. The edge
// kernel is one wave (32 lanes) per edge, 2 features per lane, with the 3x3
// attention done via wave32 shuffle reductions, then f32 atomic scatter-mean.
// ---------------------------------------------------------------------------

typedef __attribute__((ext_vector_type(2))) float v2f;
typedef __attribute__((ext_vector_type(8))) float v8f;

#define N_NODES   50000
#define N_EDGES   800000
#define IN_FEATS  128
#define OUT_FEATS 64
#define E_RAW     11
#define LN_EPS    1e-5f

// workspace layout (float offsets)
#define WS_WCQ  ((size_t)0)                         // 64*128 folded Wq@W_lin
#define WS_WCK  (WS_WCQ + 64*128)
#define WS_WCV  (WS_WCK + 64*128)
#define WS_WEQ  (WS_WCV + 64*128)                   // 64*11 folded Wq@W_e
#define WS_WEK  (WS_WEQ + 64*11)
#define WS_WEV  (WS_WEK + 64*11)
#define WS_ZQ   (WS_WEV + 64*11)                    // 50000*64
#define WS_ZK   (WS_ZQ + (size_t)N_NODES*64)
#define WS_ZV   (WS_ZK + (size_t)N_NODES*64)
#define WS_CNT  (WS_ZV + (size_t)N_NODES*64)        // 50000
#define WS_END  (WS_CNT + N_NODES)

// ---------------------------------------------------------------------------
// 0) zero the atomic accumulators (agg lives in d_out, cnt in ws)
// ---------------------------------------------------------------------------
__global__ void k_zero(float* __restrict__ agg, float* __restrict__ cnt) {
    size_t i = (size_t)blockIdx.x * blockDim.x + threadIdx.x;
    const size_t nagg = (size_t)N_NODES * OUT_FEATS;
    if (i < nagg)            agg[i] = 0.0f;
    else if (i < nagg + N_NODES) cnt[i - nagg] = 0.0f;
}

// ---------------------------------------------------------------------------
// 1) fold weights:  Wc* = W{q,k,v} @ W_lin  [64,128];  We* = W{q,k,v} @ W_e [64,11]
//    (tiny: 1.6M MACs total)
// ---------------------------------------------------------------------------
__global__ void k_fold_weights(const float* __restrict__ W_lin,  // [64,128]
                               const float* __restrict__ W_e,    // [64,11]
                               const float* __restrict__ Wq,     // [64,64]
                               const float* __restrict__ Wk,
                               const float* __restrict__ Wv,
                               float* __restrict__ ws) {
    int t = blockIdx.x * blockDim.x + threadIdx.x;
    const float* Wm[3] = {Wq, Wk, Wv};
    const int nc = 64 * 128;          // per folded node matrix
    const int ne = 64 * 11;           // per folded edge matrix
    if (t < 3 * nc) {
        int mat = t / nc, rem = t - mat * nc;
        int o = rem / 128, k = rem - o * 128;
        const float* A = Wm[mat];
        float acc = 0.0f;
        #pragma unroll 8
        for (int j = 0; j < 64; ++j) acc += A[o * 64 + j] * W_lin[j * 128 + k];
        ws[WS_WCQ + (size_t)mat * nc + rem] = acc;
    } else if (t < 3 * nc + 3 * ne) {
        int t2 = t - 3 * nc;
        int mat = t2 / ne, rem = t2 - mat * ne;
        int o = rem / 11, r = rem - o * 11;
        const float* A = Wm[mat];
        float acc = 0.0f;
        #pragma unroll 8
        for (int j = 0; j < 64; ++j) acc += A[o * 64 + j] * W_e[j * 11 + r];
        ws[WS_WEQ + (size_t)mat * ne + rem] = acc;
    }
}

// ---------------------------------------------------------------------------
// 2) node projections via WMMA fp32, q/k/v fused (A fragments loaded once,
//    fed to 3 WMMAs -> h is streamed exactly once from HBM):
//    z*[n, o] = sum_k h[n,k] * Wc*[o,k]      (N=50000, K=128, O=64)
//    grid = 3125 node tiles; 4 waves/block = 4 feature tiles; 96 WMMA/wave.
//    Fragment layouts per CDNA5 ISA 7.12.2:
//      A (16x4 f32): lanes 0-15 -> M=lane, {v0,v1}={K0,K1}; lanes 16-31 -> {K2,K3}
//      B (4x16 f32): lanes 0-15 -> N=lane, {v0,v1}={K0,K1}; lanes 16-31 -> {K2,K3}
//      C (16x16 f32): VGPR r, lane L -> row r + 8*(L>>4), col L&15
// ---------------------------------------------------------------------------
__global__ void k_node_proj_wmma(const float* __restrict__ h,   // [N,128]
                                 const float* __restrict__ ws,
                                 float* __restrict__ ws_out) {
    const int lane  = threadIdx.x & 31;
    const int wave  = threadIdx.x >> 5;      // feature tile 0..3
    const int m     = lane & 15;
    const int kh    = lane >> 4;             // selects K pair {0,1} vs {2,3}
    const int node0 = blockIdx.x * 16;
    const int feat0 = wave * 16;

    const float* arow = h + (size_t)(node0 + m) * 128 + kh * 2;
    const size_t boff = (size_t)(feat0 + m) * 128 + kh * 2;
    const float* brq  = ws + WS_WCQ + boff;
    const float* brk  = ws + WS_WCK + boff;
    const float* brv  = ws + WS_WCV + boff;

    v8f accq = {}, acck = {}, accv = {};
    #pragma unroll 4
    for (int kk = 0; kk < 32; ++kk) {        // K = 128 = 32 * 4
        v2f a  = *(const v2f*)(arow + kk * 4);
        v2f bq = *(const v2f*)(brq  + kk * 4);
        v2f bk = *(const v2f*)(brk  + kk * 4);
        v2f bv = *(const v2f*)(brv  + kk * 4);
        accq = __builtin_amdgcn_wmma_f32_16x16x4_f32(false, a, false, bq,
                                                     (short)0, accq, false, false);
        acck = __builtin_amdgcn_wmma_f32_16x16x4_f32(false, a, false, bk,
                                                     (short)0, acck, false, false);
        accv = __builtin_amdgcn_wmma_f32_16x16x4_f32(false, a, false, bv,
                                                     (short)0, accv, false, false);
    }

    const size_t ooff = (size_t)(node0 + kh * 8) * 64 + feat0 + m;
    float* oq = ws_out + WS_ZQ + ooff;
    float* ok = ws_out + WS_ZK + ooff;
    float* ov = ws_out + WS_ZV + ooff;
    #pragma unroll
    for (int r = 0; r < 8; ++r) {
        oq[(size_t)r * 64] = accq[r];
        ok[(size_t)r * 64] = acck[r];
        ov[(size_t)r * 64] = accv[r];
    }
}

// ---------------------------------------------------------------------------
// 3) fused edge kernel: one wave per edge, 2 features per lane.
// ---------------------------------------------------------------------------
__device__ __forceinline__ float wave_sum32(float p) {
    #pragma unroll
    for (int m2 = 16; m2 >= 1; m2 >>= 1) p += __shfl_xor(p, m2, 32);
    return p;
}

__global__ void k_edge_attn(const float* __restrict__ e_f,   // [E,11]
                            const int*   __restrict__ src,
                            const int*   __restrict__ dst,
                            const float* __restrict__ ws,
                            float* __restrict__ agg,          // [N,64] (= d_out)
                            float* __restrict__ cnt) {        // [N]
    const int lane = threadIdx.x & 31;
    const int e    = blockIdx.x * 8 + (threadIdx.x >> 5);
    const int s  = src[e];
    const int d  = dst[e];
    const int f  = lane * 2;

    // gathers of per-node projections (rows are 256B; fully coalesced, L2-hot)
    v2f qs = *(const v2f*)(ws + WS_ZQ + (size_t)s * 64 + f);
    v2f qd = *(const v2f*)(ws + WS_ZQ + (size_t)d * 64 + f);
    v2f ks = *(const v2f*)(ws + WS_ZK + (size_t)s * 64 + f);
    v2f kd = *(const v2f*)(ws + WS_ZK + (size_t)d * 64 + f);
    v2f vs = *(const v2f*)(ws + WS_ZV + (size_t)s * 64 + f);
    v2f vd = *(const v2f*)(ws + WS_ZV + (size_t)d * 64 + f);

    // in-register edge-token projection: [11] -> 2 features per lane, x3 mats
    float ef[E_RAW];
    #pragma unroll
    for (int j = 0; j < E_RAW; ++j) ef[j] = e_f[(size_t)e * E_RAW + j];

    const float* wq = ws + WS_WEQ + (size_t)f * E_RAW;
    const float* wk = ws + WS_WEK + (size_t)f * E_RAW;
    const float* wv = ws + WS_WEV + (size_t)f * E_RAW;
    v2f qe = {}, ke = {}, ve = {};
    #pragma unroll
    for (int j = 0; j < E_RAW; ++j) {
        float c = ef[j];
        qe.x += wq[j] * c;         qe.y += wq[E_RAW + j] * c;
        ke.x += wk[j] * c;         ke.y += wk[E_RAW + j] * c;
        ve.x += wv[j] * c;         ve.y += wv[E_RAW + j] * c;
    }

    v2f Q[3] = {qs, qd, qe};
    v2f K[3] = {ks, kd, ke};
    v2f V[3] = {vs, vd, ve};

    // S[i][j] = Q_i . K_j over all 64 features (wave32 reduction)
    float S[3][3];
    #pragma unroll
    for (int i = 0; i < 3; ++i)
        #pragma unroll
        for (int j = 0; j < 3; ++j)
            S[i][j] = wave_sum32(Q[i].x * K[j].x + Q[i].y * K[j].y);

    // softmax over j, then column weights w_j = sum_i attn[i][j]
    float colw[3] = {0.f, 0.f, 0.f};
    #pragma unroll
    for (int i = 0; i < 3; ++i) {
        float mx = fmaxf(S[i][0], fmaxf(S[i][1], S[i][2]));
        float e0 = __expf(S[i][0] - mx);
        float e1 = __expf(S[i][1] - mx);
        float e2 = __expf(S[i][2] - mx);
        float inv = 1.0f / (e0 + e1 + e2);
        colw[0] += e0 * inv; colw[1] += e1 * inv; colw[2] += e2 * inv;
    }

    v2f m2;
    m2.x = colw[0] * V[0].x + colw[1] * V[1].x + colw[2] * V[2].x;
    m2.y = colw[0] * V[0].y + colw[1] * V[1].y + colw[2] * V[2].y;

    atomicAdd(&agg[(size_t)d * 64 + f],     m2.x);
    atomicAdd(&agg[(size_t)d * 64 + f + 1], m2.y);
    if (lane == 0) atomicAdd(&cnt[d], 1.0f);
}

// ---------------------------------------------------------------------------
// 4) finalize: scatter-mean divide, ReLU, LayerNorm. one wave per node.
//    agg == d_out, updated in place (each wave owns its whole row).
// ---------------------------------------------------------------------------
__global__ void k_finalize(float* __restrict__ agg,           // [N,64]
                           const float* __restrict__ cnt,
                           const float* __restrict__ gamma,
                           const float* __restrict__ beta) {
    const int lane = threadIdx.x & 31;
    const int n    = blockIdx.x * 8 + (threadIdx.x >> 5);
    const int f    = lane * 2;

    v2f a = *(const v2f*)(agg + (size_t)n * 64 + f);
    float inv_c = 1.0f / fmaxf(cnt[n], 1.0f);
    float r0 = fmaxf(a.x * inv_c, 0.0f);
    float r1 = fmaxf(a.y * inv_c, 0.0f);

    float s1 = r0 + r1;
    float s2 = r0 * r0 + r1 * r1;
    #pragma unroll
    for (int m2 = 16; m2 >= 1; m2 >>= 1) {
        s1 += __shfl_xor(s1, m2, 32);
        s2 += __shfl_xor(s2, m2, 32);
    }
    float mu  = s1 * (1.0f / 64.0f);
    float var = s2 * (1.0f / 64.0f) - mu * mu;
    float is  = rsqrtf(var + LN_EPS);

    v2f o;
    o.x = (r0 - mu) * is * gamma[f]     + beta[f];
    o.y = (r1 - mu) * is * gamma[f + 1] + beta[f + 1];
    *(v2f*)(agg + (size_t)n * 64 + f) = o;
}

// ---------------------------------------------------------------------------
extern "C" void kernel_launch(void* const* d_in, const int* in_sizes, int n_in,
                              void* d_out, int out_size, void* d_ws, size_t ws_size,
                              hipStream_t stream) {
    const float* h     = (const float*)d_in[0];
    const float* e_f   = (const float*)d_in[1];
    const int*   src   = (const int*)  d_in[2];
    const int*   dst   = (const int*)  d_in[3];
    const float* W_lin = (const float*)d_in[4];
    const float* W_e   = (const float*)d_in[5];
    const float* Wq    = (const float*)d_in[6];
    const float* Wk    = (const float*)d_in[7];
    const float* Wv    = (const float*)d_in[8];
    const float* gamma = (const float*)d_in[9];
    const float* beta  = (const float*)d_in[10];

    float* ws  = (float*)d_ws;   // needs WS_END*4 ~= 38.8 MB
    float* out = (float*)d_out;  // [50000,64] f32; doubles as scatter accumulator

    // 0) zero accumulators
    {
        size_t n = (size_t)N_NODES * OUT_FEATS + N_NODES;
        k_zero<<<(unsigned)((n + 255) / 256), 256, 0, stream>>>(out, ws + WS_CNT);
    }
    // 1) fold weights
    {
        int n = 3 * 64 * 128 + 3 * 64 * 11;
        k_fold_weights<<<(n + 255) / 256, 256, 0, stream>>>(W_lin, W_e, Wq, Wk, Wv, ws);
    }
    // 2) node projections (WMMA, q/k/v fused): grid = 50000/16 node tiles
    k_node_proj_wmma<<<N_NODES / 16, 128, 0, stream>>>(h, ws, ws);
    // 3) fused per-edge attention + scatter (wave per edge, 8 waves/block)
    k_edge_attn<<<N_EDGES / 8, 256, 0, stream>>>(e_f, src, dst, ws, out, ws + WS_CNT);
    // 4) mean + ReLU + LayerNorm in place
    k_finalize<<<N_NODES / 8, 256, 0, stream>>>(out, ws + WS_CNT, gamma, beta);
}